// SimpleModel_19963007991948
// MI455X (gfx1250) — compile-verified
//
#include <hip/hip_runtime.h>

typedef __attribute__((ext_vector_type(2))) float v2f;
typedef __attribute__((ext_vector_type(8))) float v8f;

#define D_IN   100   // model dim
#define NPAD   112   // 100 padded up to 7 * 16 (output columns)
#define KPAD   102   // K stride for transposed M: even (8B align) and 38 mod 64 (bank-conflict-free)
#define NTILES 7     // column tiles of 16
#define KSTEPS 25    // 100 / 4
#define MT_ELEMS (NPAD * KPAD)   // 11424 floats = 45.7 KB

// ---------------------------------------------------------------------------
// Prep: Mt[f][d] = M[d][f] = sum_e W[e][d] * S[e][f]   (M = W^T @ states),
// stored transposed [NPAD][KPAD] with zero padding; cp[f] = sum_e b[e]*S[e][f].
// blockIdx 0..111 -> rows of Mt (output column f); blockIdx 112 -> bias row.
// ---------------------------------------------------------------------------
__global__ void prep_Mt_kernel(const float* __restrict__ W,
                               const float* __restrict__ b,
                               const float* __restrict__ S,
                               float* __restrict__ Mt,   // [NPAD][KPAD]
                               float* __restrict__ cp) { // [NPAD]
    const int d = threadIdx.x;   // K index (0..KPAD-1), blockDim = 128
    const int f = blockIdx.x;    // output column (0..NPAD) ; NPAD = bias row
    if (f < NPAD) {
        if (d < KPAD) {
            float acc = 0.0f;
            if (f < D_IN && d < D_IN) {
                for (int e = 0; e < D_IN; ++e)
                    acc += W[e * D_IN + d] * S[e * D_IN + f];
            }
            Mt[f * KPAD + d] = acc;          // padding lanes write 0
        }
    } else {
        if (d < NPAD) {
            float acc = 0.0f;
            if (d < D_IN) {
                for (int e = 0; e < D_IN; ++e)
                    acc += b[e] * S[e * D_IN + d];
            }
            cp[d] = acc;
        }
    }
}

// ---------------------------------------------------------------------------
// Main GEMM: Y[N,100] = X[N,100] @ M + c via V_WMMA_F32_16X16X4_F32.
// M lives in LDS (transposed); X streams from HBM once; one wave = 16x112 tile.
// ---------------------------------------------------------------------------
__global__ void __launch_bounds__(256)
gemm_wmma_f32_kernel(const float* __restrict__ X,
                     const float* __restrict__ Mt,
                     const float* __restrict__ cp,
                     float* __restrict__ Y) {
    __shared__ float sMt[MT_ELEMS];

    // Cooperative, fully coalesced copy of Mt into LDS (45.7 KB).
    for (int i = threadIdx.x; i < MT_ELEMS; i += 256)
        sMt[i] = Mt[i];
    __syncthreads();

    const int lane = threadIdx.x & 31;
    const int wave = threadIdx.x >> 5;
    const int tile = blockIdx.x * 8 + wave;     // 16-row tile index
    const int row0 = tile * 16;

    const int r    = lane & 15;                 // A row / B,D column within tile
    const int klo  = (lane >> 4) << 1;          // 0 for lanes 0-15, 2 for lanes 16-31

    const float* __restrict__ xrow = X + (size_t)(row0 + r) * D_IN;

    v8f acc[NTILES] = {};

    for (int kk = 0; kk < KSTEPS; ++kk) {
        const int k = kk * 4 + klo;             // this half-wave's K pair: k, k+1
        // A fragment: X[row0+r][k..k+1], 8B-aligned global load (only HBM stream).
        v2f a = *(const v2f*)(xrow + k);
#pragma unroll
        for (int t = 0; t < NTILES; ++t) {
            // B fragment: Mt[col][k..k+1] -> single 8B LDS read, bank-conflict-free.
            v2f bf = *(const v2f*)(sMt + (t * 16 + r) * KPAD + k);
            acc[t] = __builtin_amdgcn_wmma_f32_16x16x4_f32(
                false, a, false, bf, (short)0, acc[t], false, false);
        }
    }

    // Bias add + store.  D layout: VGPR i -> row (i or i+8 by half-wave), col = t*16 + r.
    const int rbase = row0 + ((lane >> 4) << 3);
#pragma unroll
    for (int t = 0; t < NTILES; ++t) {
        const int col = t * 16 + r;
        if (col < D_IN) {
            const float cb = cp[col];
#pragma unroll
            for (int i = 0; i < 8; ++i) {
                Y[(size_t)(rbase + i) * D_IN + col] = acc[t][i] + cb;
            }
        }
    }
}

// ---------------------------------------------------------------------------
extern "C" void kernel_launch(void* const* d_in, const int* in_sizes, int n_in,
                              void* d_out, int out_size, void* d_ws, size_t ws_size,
                              hipStream_t stream) {
    const float* x      = (const float*)d_in[0];  // [4096,64,1,100]
    const float* W      = (const float*)d_in[1];  // [100,100]
    const float* b      = (const float*)d_in[2];  // [100]
    const float* states = (const float*)d_in[3];  // [100,100]
    float*       out    = (float*)d_out;          // [4096,64,1,100]

    float* Mt = (float*)d_ws;                     // [NPAD][KPAD] transposed fused matrix
    float* cp = Mt + MT_ELEMS;                    // [NPAD] fused bias

    // Build fused matrix M = W^T @ states (transposed, zero-padded) and c = b @ states.
    prep_Mt_kernel<<<dim3(NPAD + 1), dim3(128), 0, stream>>>(W, b, states, Mt, cp);

    // N = 4096*64 = 262144 rows; 16 rows/wave * 8 waves/block = 128 rows/block.
    const int nrows  = 4096 * 64;
    const int blocks = nrows / 128;               // 2048, exact
    gemm_wmma_f32_kernel<<<dim3(blocks), dim3(256), 0, stream>>>(x, Mt, cp, out);
}